// FeatureLabelLoss_57853209477365
// MI455X (gfx1250) — compile-verified
//
#include <hip/hip_runtime.h>
#include <math.h>

// CDNA5 / gfx1250: wave32, WMMA f32 16x16x4.
typedef __attribute__((ext_vector_type(2))) float v2f;
typedef __attribute__((ext_vector_type(8))) float v8f;

#define B_N     32
#define C_N     5000
#define D_N     512
#define TILES_N ((C_N + 15) / 16)   // 313 c-tiles of 16
#define NPART   (B_N * TILES_N)     // 10016 partial sums

// One wave (32 threads) per (b, c-tile of 16).
// acc(16x16) += A(16x4) * B(4x16) over D, where
//   A[m][k] = features[b][c0+m][d0+k],  B[k][n] = embeddings[c0+n][d0+k]
// so diag(acc)[m] = dot(features[b][c0+m], embeddings[c0+m]) in exact f32.
//
// ISA A-layout (16x4 f32): lane L holds row (L&15), K = {0,1} (lanes 0-15)
// or K = {2,3} (lanes 16-31) -> one contiguous global_load_b64 per operand
// per k-step. The 4x16 B-layout is lane-for-lane identical in addressing.
// ||f||^2 / ||e||^2 accumulate as packed v_pk_fma_f32 on the same registers
// (lanes r and r+16 hold complementary K-halves of row r).
// Two WMMA accumulators (even/odd k-steps) break the serial D->C chain.
__global__ __launch_bounds__(32)
void fll_sim_kernel(const float* __restrict__ feat,
                    const float* __restrict__ emb,
                    const float* __restrict__ lab,
                    float* __restrict__ partial)
{
    const int tile = blockIdx.x;
    const int b    = blockIdx.y;
    const int lane = threadIdx.x;        // 0..31, one full wave32
    const int row  = lane & 15;
    const int koff = (lane >> 4) << 1;   // 0 (lanes 0-15) or 2 (lanes 16-31)

    int c = tile * 16 + row;
    c = (c < C_N) ? c : (C_N - 1);       // clamp tail (EXEC must stay all-1s)

    const float* fptr = feat + ((size_t)b * C_N + (size_t)c) * D_N + koff;
    const float* eptr = emb  + (size_t)c * D_N + koff;

    v8f acc0 = {}, acc1 = {};
    v2f nf2v = {}, ne2v = {};            // packed norm accumulators

#pragma unroll 8
    for (int d = 0; d < D_N; d += 8) {
        v2f a0 = *(const v2f*)(fptr + d);       // global_load_b64
        v2f b0 = *(const v2f*)(eptr + d);       // global_load_b64 (L2-hot)
        v2f a1 = *(const v2f*)(fptr + d + 4);
        v2f b1 = *(const v2f*)(eptr + d + 4);
        acc0 = __builtin_amdgcn_wmma_f32_16x16x4_f32(
                   false, a0, false, b0, (short)0, acc0, false, false);
        acc1 = __builtin_amdgcn_wmma_f32_16x16x4_f32(
                   false, a1, false, b1, (short)0, acc1, false, false);
        nf2v += a0 * a0;                        // v_pk_fma_f32, no moves
        ne2v += b0 * b0;
        nf2v += a1 * a1;
        ne2v += b1 * b1;
    }

    v8f acc = acc0 + acc1;
    float nf2 = nf2v.x + nf2v.y;
    float ne2 = ne2v.x + ne2v.y;

    // Extract diagonal. C/D layout: VGPR v holds M=v (lanes 0-15, N=lane)
    // and M=v+8 (lanes 16-31, N=lane-16). So diag m<8 lives in lane m
    // (component m); diag m>=8 lives in lane m+16 (component m-8).
    const int idx = lane & 7;
    float e0 = (idx & 1) ? acc.s1 : acc.s0;
    float e1 = (idx & 1) ? acc.s3 : acc.s2;
    float e2 = (idx & 1) ? acc.s5 : acc.s4;
    float e3 = (idx & 1) ? acc.s7 : acc.s6;
    float f0 = (idx & 2) ? e1 : e0;
    float f1 = (idx & 2) ? e3 : e2;
    float dval = (idx & 4) ? f1 : f0;

    const int src = (lane < 8) ? lane : (lane + 8);
    float simraw = __shfl(dval, src, 32);               // valid on lanes 0-15
    float nf2t   = nf2 + __shfl(nf2, lane | 16, 32);    // row (L&15) full sum
    float ne2t   = ne2 + __shfl(ne2, lane | 16, 32);

    // Loss term for (b, c0+lane), lanes 0-15 only; tail c's masked out.
    float lsum = 0.0f;
    const int cc = tile * 16 + (lane & 15);
    if (lane < 16 && cc < C_N) {
        float nf = fmaxf(sqrtf(nf2t), 1e-12f);          // F.normalize eps
        float ne = fmaxf(sqrtf(ne2t), 1e-12f);
        float s  = simraw / (nf * ne);
        float L  = lab[(size_t)b * C_N + cc];
        float Sv  = (1.0f + s) * 0.5f + 1e-6f;
        float pos = L * logf(Sv);
        float inner = 1.0f
            - ((float)(C_N - 1) / (float)C_N)
              * fabsf(1.0f / (float)(C_N - 1) + s)
            + 1e-6f;
        float neg = (1.0f - L) * logf(inner);
        lsum = pos + neg;
    }

    // Fixed-order wave reduction -> one partial per block (deterministic).
#pragma unroll
    for (int off = 16; off > 0; off >>= 1)
        lsum += __shfl_xor(lsum, off, 32);
    if (lane == 0)
        partial[(size_t)b * TILES_N + tile] = lsum;
}

// Deterministic final reduction: fixed strided order per lane, then a
// fixed-order butterfly. Writes d_out[0] with '=' (no init needed).
__global__ __launch_bounds__(32)
void fll_reduce_kernel(const float* __restrict__ partial,
                       float* __restrict__ out)
{
    const int lane = threadIdx.x;
    float s = 0.0f;
    for (int i = lane; i < NPART; i += 32)
        s += partial[i];
#pragma unroll
    for (int off = 16; off > 0; off >>= 1)
        s += __shfl_xor(s, off, 32);
    if (lane == 0)
        out[0] = -s / (float)(B_N * C_N);
}

extern "C" void kernel_launch(void* const* d_in, const int* in_sizes, int n_in,
                              void* d_out, int out_size, void* d_ws, size_t ws_size,
                              hipStream_t stream) {
    (void)in_sizes; (void)n_in; (void)out_size; (void)ws_size;
    const float* feat = (const float*)d_in[0];   // (32, 5000, 512) f32
    const float* emb  = (const float*)d_in[1];   // (5000, 512)     f32
    const float* lab  = (const float*)d_in[2];   // (32, 5000)      f32
    float* out        = (float*)d_out;           // scalar f32
    float* partial    = (float*)d_ws;            // NPART floats (~40 KB)

    dim3 grid(TILES_N, B_N);                     // 313 x 32 waves
    fll_sim_kernel<<<grid, 32, 0, stream>>>(feat, emb, lab, partial);
    fll_reduce_kernel<<<1, 32, 0, stream>>>(partial, out);
}